// ResiualVectorQuantizer_3796751089900
// MI455X (gfx1250) — compile-verified
//
#include <hip/hip_runtime.h>
#include <hip/hip_bf16.h>

// RVQ (Encodec residual vector quantizer) for MI455X / gfx1250.
// Shapes: x (B=8, D=128, T=4096) f32; codebooks (NQ=8, K=1024, D=128) f32.
// Outputs concatenated in d_out (float): quantized (B,D,T) | codes (NQ,B,T) | bw | penalty.
//
// Per stage: dist = ||c||^2 - 2*(R @ C^T) via V_WMMA_F32_16X16X4_F32 (full fp32
// so the argmin matches the fp32 reference). Codebook N-tiles are streamed
// global->LDS with GLOBAL_LOAD_ASYNC_TO_LDS_B128 (ASYNCcnt) into a double
// buffer so the next tile's DMA overlaps the current tile's 32 chained WMMAs.
// quantized = x - final residual; commit_s = mean(residual_new^2).

#define RVQ_B   8
#define RVQ_D   128
#define RVQ_T   4096
#define RVQ_BT  (RVQ_B * RVQ_T)      // 32768
#define RVQ_NQ  8
#define RVQ_K   1024
#define LDS_STRIDE 132               // pad 128 -> 132 floats: conflict-free f2 reads

typedef __attribute__((ext_vector_type(2))) float v2f;
typedef __attribute__((ext_vector_type(8))) float v8f;

// ---------------------------------------------------------------------------
// 1) Transpose x (B,D,T) -> resid (B*T, D), LDS-tiled, coalesced both ways.
__global__ __launch_bounds__(256) void rvq_transpose_in(
    const float* __restrict__ x, float* __restrict__ resid) {
  __shared__ float tile[32][33];
  const int b  = blockIdx.z;
  const int t0 = blockIdx.x * 32;
  const int d0 = blockIdx.y * 32;
  const int tx = threadIdx.x, ty = threadIdx.y;  // (32, 8)
#pragma unroll
  for (int i = 0; i < 4; ++i)  // read along T (coalesced)
    tile[ty + 8 * i][tx] =
        x[(size_t)(b * RVQ_D + d0 + ty + 8 * i) * RVQ_T + t0 + tx];
  __syncthreads();
#pragma unroll
  for (int i = 0; i < 4; ++i)  // write along D (coalesced)
    resid[(size_t)(b * RVQ_T + t0 + ty + 8 * i) * RVQ_D + d0 + tx] =
        tile[tx][ty + 8 * i];
}

// ---------------------------------------------------------------------------
// 2) Codebook row norms + zero commit accumulators.
__global__ __launch_bounds__(256) void rvq_norms(
    const float* __restrict__ cb, float* __restrict__ cbnorm,
    float* __restrict__ commit) {
  const int row = blockIdx.x * blockDim.x + threadIdx.x;  // 0 .. NQ*K-1
  if (row < RVQ_NQ * RVQ_K) {
    const float* p = cb + (size_t)row * RVQ_D;
    float s = 0.f;
#pragma unroll 8
    for (int d = 0; d < RVQ_D; ++d) s += p[d] * p[d];
    cbnorm[row] = s;
  }
  if (blockIdx.x == 0 && threadIdx.x < RVQ_NQ) commit[threadIdx.x] = 0.f;
}

// ---------------------------------------------------------------------------
// 3) One RVQ stage. Block = 256 threads = 8 waves; each wave owns 16 rows.
//    Codebook tiles double-buffered in LDS via async global->LDS DMA.
__global__ __launch_bounds__(256) void rvq_stage(
    float* __restrict__ resid, const float* __restrict__ cb,
    const float* __restrict__ cbnorm, float* __restrict__ codes,
    float* __restrict__ commit) {
  __shared__ __align__(16) float cbtile[2][16 * LDS_STRIDE];
  const int tid  = threadIdx.x;
  const int lane = tid & 31;
  const int wave = tid >> 5;
  const int half = lane >> 4;   // 0: lanes 0-15, 1: lanes 16-31
  const int l16  = lane & 15;
  const int pbase = blockIdx.x * 128 + wave * 16;  // first residual row

  // A fragments: residual tile 16xD held in registers for the whole stage.
  // ISA layout (32-bit A 16x4): lane = m + 16*half holds K = 4k+2*half+{0,1}.
  v2f afrag[32];
  {
    const float* arow = resid + (size_t)(pbase + l16) * RVQ_D + 2 * half;
#pragma unroll
    for (int kb = 0; kb < 32; ++kb)
      afrag[kb] = *(const v2f*)(arow + 4 * kb);
  }

  float bestv[8];
  int   besti[8];
#pragma unroll
  for (int r = 0; r < 8; ++r) { bestv[r] = 3.4e38f; besti[r] = 0; }

  const int lrow = tid >> 5;         // 0..7  (cooperative tile fill)
  const int lcol = (tid & 31) * 4;   // 0,4,...,124

  // Async global->LDS: each lane moves 16B; 2 issues/thread = 8KB tile/block.
  // VDST VGPR carries the LDS byte address (low 32 bits of the flat shared
  // pointer encode the LDS offset); tracked with ASYNCcnt.
  auto issue_tile = [&](int nt, int buf) {
    const float* g0 = cb + (size_t)(nt * 16 + lrow) * RVQ_D + lcol;
    const float* g1 = g0 + 8 * RVQ_D;
    unsigned l0 = (unsigned)(size_t)&cbtile[buf][lrow * LDS_STRIDE + lcol];
    unsigned l1 = (unsigned)(size_t)&cbtile[buf][(lrow + 8) * LDS_STRIDE + lcol];
    asm volatile(
        "global_load_async_to_lds_b128 %0, %1, off\n\t"
        "global_load_async_to_lds_b128 %2, %3, off"
        :: "v"(l0), "v"(g0), "v"(l1), "v"(g1)
        : "memory");
  };

  issue_tile(0, 0);

  for (int nt = 0; nt < RVQ_K / 16; ++nt) {
    const int cur = nt & 1;
    // My async loads for buf[cur] have landed in LDS:
    asm volatile("s_wait_asynccnt 0x0" ::: "memory");
    // Everyone's loads for buf[cur] landed; everyone's reads of buf[cur^1]
    // (previous iteration) are already in VGPRs (waited before their WMMAs):
    __syncthreads();
    if (nt + 1 < RVQ_K / 16) issue_tile(nt + 1, cur ^ 1);

    // S[m][n] = sum_d R[m][d] * C[n][d]; B[k][n] = C[n][k], mirrored layout.
    const float* tileP = &cbtile[cur][0];
    v8f acc = {};
#pragma unroll
    for (int kb = 0; kb < 32; ++kb) {
      v2f bfrag = *(const v2f*)&tileP[l16 * LDS_STRIDE + 4 * kb + 2 * half];
      acc = __builtin_amdgcn_wmma_f32_16x16x4_f32(
          false, afrag[kb], false, bfrag, (short)0, acc, false, false);
    }

    // dist = ||c_n||^2 - 2*S ; lane's column n = nt*16 + l16 for all 8 rows.
    const int   nidx = nt * 16 + l16;
    const float nrm  = cbnorm[nidx];
#pragma unroll
    for (int r = 0; r < 8; ++r) {
      float dv = nrm - 2.0f * acc[r];       // acc[r] = S[r + 8*half][l16]
      if (dv < bestv[r]) { bestv[r] = dv; besti[r] = nidx; }
    }
  }

  // Argmin across the 16 lanes of each half (halves carry disjoint M rows).
  // xor masks 1..8 never toggle bit4, so halves reduce independently.
#pragma unroll
  for (int r = 0; r < 8; ++r) {
#pragma unroll
    for (int mask = 1; mask <= 8; mask <<= 1) {
      float ov = __shfl_xor(bestv[r], mask, 32);
      int   oi = __shfl_xor(besti[r], mask, 32);
      if (ov < bestv[r] || (ov == bestv[r] && oi < besti[r])) {
        bestv[r] = ov; besti[r] = oi;
      }
    }
  }

  // Update residual rows, emit codes, accumulate commit (= sum r_new^2).
  float sq = 0.f;
#pragma unroll
  for (int m = 0; m < 16; ++m) {
    const int idx = __shfl(besti[m & 7], (m >> 3) * 16, 32);  // broadcast
    const float* q   = cb + (size_t)idx * RVQ_D;
    float*       rrw = resid + (size_t)(pbase + m) * RVQ_D;
    float rn0 = rrw[lane]      - q[lane];
    float rn1 = rrw[lane + 32] - q[lane + 32];
    float rn2 = rrw[lane + 64] - q[lane + 64];
    float rn3 = rrw[lane + 96] - q[lane + 96];
    rrw[lane]      = rn0;
    rrw[lane + 32] = rn1;
    rrw[lane + 64] = rn2;
    rrw[lane + 96] = rn3;
    sq += rn0 * rn0 + rn1 * rn1 + rn2 * rn2 + rn3 * rn3;
    if (lane == 0) codes[pbase + m] = (float)idx;
  }
#pragma unroll
  for (int mask = 16; mask >= 1; mask >>= 1) sq += __shfl_xor(sq, mask, 32);
  if (lane == 0) atomicAdd(commit, sq);
}

// ---------------------------------------------------------------------------
// 4) quantized = x - final residual, transposed back to (B,D,T).
__global__ __launch_bounds__(256) void rvq_finalize_out(
    const float* __restrict__ x, const float* __restrict__ resid,
    float* __restrict__ outq) {
  __shared__ float tile[32][33];
  const int b  = blockIdx.z;
  const int t0 = blockIdx.x * 32;
  const int d0 = blockIdx.y * 32;
  const int tx = threadIdx.x, ty = threadIdx.y;
#pragma unroll
  for (int i = 0; i < 4; ++i)  // read resid along D (coalesced)
    tile[ty + 8 * i][tx] =
        resid[(size_t)(b * RVQ_T + t0 + ty + 8 * i) * RVQ_D + d0 + tx];
  __syncthreads();
#pragma unroll
  for (int i = 0; i < 4; ++i) {  // write along T (coalesced)
    size_t o = (size_t)(b * RVQ_D + d0 + ty + 8 * i) * RVQ_T + t0 + tx;
    outq[o] = x[o] - tile[tx][ty + 8 * i];
  }
}

// 5) bw + penalty scalars.
__global__ void rvq_scalars(const float* __restrict__ commit,
                            const int* __restrict__ frame_rate,
                            float* __restrict__ outs) {
  if (blockIdx.x == 0 && threadIdx.x == 0) {
    float s = 0.f;
    for (int i = 0; i < RVQ_NQ; ++i) s += commit[i];
    outs[0] = (float)RVQ_NQ * 10.0f * (float)frame_rate[0];  // log2(1024)=10
    outs[1] = s / ((float)RVQ_NQ * (float)RVQ_BT * (float)RVQ_D);
  }
}

// ---------------------------------------------------------------------------
extern "C" void kernel_launch(void* const* d_in, const int* in_sizes, int n_in,
                              void* d_out, int out_size, void* d_ws, size_t ws_size,
                              hipStream_t stream) {
  const float* x  = (const float*)d_in[0];                 // (8,128,4096)
  const float* cb = (const float*)d_in[1];                 // (8,1024,128)
  const int*   fr = (const int*)d_in[2];                   // frame_rate

  // Workspace layout (bytes): resid 16 MB | cbnorm 32 KB | commit 32 B
  char* ws = (char*)d_ws;
  float* resid  = (float*)ws;                                      // BT*D
  float* cbnorm = (float*)(ws + (size_t)RVQ_BT * RVQ_D * 4);       // NQ*K
  float* commit = (float*)(ws + (size_t)RVQ_BT * RVQ_D * 4 +
                           (size_t)RVQ_NQ * RVQ_K * 4);            // NQ

  float* out_quant = (float*)d_out;                                // B*D*T
  float* out_codes = out_quant + (size_t)RVQ_B * RVQ_D * RVQ_T;    // NQ*B*T
  float* out_scal  = out_codes + (size_t)RVQ_NQ * RVQ_B * RVQ_T;   // bw, penalty

  // 1) residual = x^T ; 2) norms + zero commit
  rvq_transpose_in<<<dim3(RVQ_T / 32, RVQ_D / 32, RVQ_B), dim3(32, 8), 0,
                     stream>>>(x, resid);
  rvq_norms<<<(RVQ_NQ * RVQ_K + 255) / 256, 256, 0, stream>>>(cb, cbnorm,
                                                              commit);

  // 3) eight sequential stages (residual carries the dependency)
  for (int s = 0; s < RVQ_NQ; ++s) {
    rvq_stage<<<RVQ_BT / 128, 256, 0, stream>>>(
        resid, cb + (size_t)s * RVQ_K * RVQ_D, cbnorm + (size_t)s * RVQ_K,
        out_codes + (size_t)s * RVQ_BT, commit + s);
  }

  // 4) quantized = x - residual ; 5) scalars
  rvq_finalize_out<<<dim3(RVQ_T / 32, RVQ_D / 32, RVQ_B), dim3(32, 8), 0,
                     stream>>>(x, resid, out_quant);
  rvq_scalars<<<1, 32, 0, stream>>>(commit, fr, out_scal);
}